// UnimolEmbedding_91053306675334
// MI455X (gfx1250) — compile-verified
//
#include <hip/hip_runtime.h>
#include <hip/hip_bf16.h>
#include <math.h>

typedef float v2f __attribute__((ext_vector_type(2)));
typedef float v8f __attribute__((ext_vector_type(8)));

#define ATOMV   128
#define APO     128
#define NHEAD   16
#define DMODEL  512
#define BSZ     4
#define LSEQ    384
#define JTILES  (LSEQ / 16)      // 24
#define WPB     8                // waves per block (256 threads, wave32)
#define INV_SQRT_2PI 0.3989422804014327f

// ---------------------------------------------------------------------------
// Kernel 1: atoms_emb[l,b,d] = atype_emb[atoms[b,l],d] + chiral_emb[chirals[b,l],d]
// Pure gather-add, float4 per thread, fully coalesced.
// ---------------------------------------------------------------------------
__global__ __launch_bounds__(DMODEL / 4)
void atoms_emb_kernel(const int* __restrict__ atoms,
                      const int* __restrict__ chirals,
                      const float* __restrict__ atype_emb,
                      const float* __restrict__ chiral_emb,
                      float* __restrict__ out) {
    const int bl = blockIdx.x;                 // b * LSEQ + l
    const int b  = bl / LSEQ;
    const int l  = bl - b * LSEQ;
    const int d  = threadIdx.x * 4;

    const int a = atoms[bl];
    const int c = chirals[bl];

    const float4 va = *reinterpret_cast<const float4*>(atype_emb  + (size_t)a * DMODEL + d);
    const float4 vc = *reinterpret_cast<const float4*>(chiral_emb + (size_t)c * DMODEL + d);
    float4 r;
    r.x = va.x + vc.x; r.y = va.y + vc.y; r.z = va.z + vc.z; r.w = va.w + vc.w;
    *reinterpret_cast<float4*>(out + ((size_t)l * BSZ + b) * DMODEL + d) = r;
}

// ---------------------------------------------------------------------------
// Kernel 2: pair bias.
// One wave = one (b, i, 16-wide j tile). Gaussian features live only in
// registers; 128->16 projection via 32x V_WMMA_F32_16X16X4_F32 (exact fp32).
//
// A (16x4 f32)  VGPR layout: lane L: m = L&15; v0: k = 4c + 2*(L>>4), v1: k+1
// B (4x16 f32)  VGPR layout: lane L: n = L&15; v0: k = 4c + 2*(L>>4), v1: k+1
// C/D (16x16)   VGPR r:      lane L: h = L&15; m = r + 8*(L>>4)
// ---------------------------------------------------------------------------
__global__ __launch_bounds__(WPB * 32)
void pair_bias_kernel(const int*   __restrict__ atoms,
                      const float* __restrict__ coords,       // [B,L,3]
                      const int*   __restrict__ bonds,        // [B,L,L]
                      const float* __restrict__ apair_w,      // [ATOMV*ATOMV, APO]
                      const float* __restrict__ apair_b,      // [ATOMV*ATOMV, APO]
                      const float* __restrict__ means,        // [APO]
                      const float* __restrict__ stds,         // [APO]
                      const float* __restrict__ bond_emb,     // [32, NHEAD]
                      const float* __restrict__ lin_w,        // [NHEAD, APO]
                      const float* __restrict__ lin_b,        // [NHEAD]
                      float* __restrict__ out_ap) {           // [B, NHEAD, L, L]
    __shared__ float smem[WPB][16 * 17];

    const int lane = threadIdx.x & 31;
    const int wv   = threadIdx.x >> 5;
    const int tile = blockIdx.x * WPB + wv;

    const int b   = tile / (LSEQ * JTILES);
    const int rem = tile - b * (LSEQ * JTILES);
    const int i   = rem / JTILES;
    const int j0  = (rem - i * JTILES) * 16;

    const int m    = lane & 15;        // A-matrix row  (j within tile)
    const int half = lane >> 4;        // k sub-slot selector
    const int j    = j0 + m;

    // --- per-lane pair scalars: distance + atom-pair table row -------------
    const int ai = atoms[b * LSEQ + i];
    const int aj = atoms[b * LSEQ + j];
    const float* ci = coords + ((size_t)b * LSEQ + i) * 3;
    const float* cj = coords + ((size_t)b * LSEQ + j) * 3;
    const float dx = ci[0] - cj[0];
    const float dy = ci[1] - cj[1];
    const float dz = ci[2] - cj[2];
    const float dist = sqrtf(dx * dx + dy * dy + dz * dz + 1e-12f);

    const int pair_idx = aj * ATOMV + ai;      // apair_idx[b,i,j] = atoms[b,j]*128 + atoms[b,i]
    const float* wrow = apair_w + (size_t)pair_idx * APO;
    const float* brow = apair_b + (size_t)pair_idx * APO;
    const float* lrow = lin_w   + (size_t)(lane & 15) * APO;   // n-th head column of B

    v8f acc = {0.f, 0.f, 0.f, 0.f, 0.f, 0.f, 0.f, 0.f};

    const int kbase = 2 * half;
    #pragma unroll
    for (int c = 0; c < APO / 4; ++c) {
        const int k0 = 4 * c + kbase;          // even -> 8B-aligned float2 loads

        const float2 wv2 = *reinterpret_cast<const float2*>(wrow  + k0);
        const float2 bv2 = *reinterpret_cast<const float2*>(brow  + k0);
        const float2 mv2 = *reinterpret_cast<const float2*>(means + k0);
        const float2 sv2 = *reinterpret_cast<const float2*>(stds  + k0);

        // single v_rcp_f32 per std (avoid full IEEE divide expansion)
        const float is0 = __builtin_amdgcn_rcpf(fabsf(sv2.x) + 1e-5f);
        const float is1 = __builtin_amdgcn_rcpf(fabsf(sv2.y) + 1e-5f);
        const float t0  = (wv2.x * dist + bv2.x - mv2.x) * is0;
        const float t1  = (wv2.y * dist + bv2.y - mv2.y) * is1;
        const float a0  = __expf(-0.5f * t0 * t0) * (INV_SQRT_2PI * is0);
        const float a1  = __expf(-0.5f * t1 * t1) * (INV_SQRT_2PI * is1);

        const float2 lw2 = *reinterpret_cast<const float2*>(lrow + k0);

        v2f Af = {a0, a1};
        v2f Bf = {lw2.x, lw2.y};
        acc = __builtin_amdgcn_wmma_f32_16x16x4_f32(
            /*neg_a=*/false, Af, /*neg_b=*/false, Bf,
            /*c_mod=*/(short)0, acc, /*reuse_a=*/false, /*reuse_b=*/false);
    }

    // --- epilogue: + lin_b + bond_emb, key-pad mask, LDS transpose ---------
    const int h   = lane & 15;                 // C/D column = head
    const float lb = lin_b[h];
    #pragma unroll
    for (int r = 0; r < 8; ++r) {
        const int mr = r + 8 * half;           // C/D row within tile
        const int jr = j0 + mr;
        const int bd = bonds[((size_t)b * LSEQ + i) * LSEQ + jr];
        float v = acc[r] + lb + bond_emb[bd * NHEAD + h];
        if (atoms[b * LSEQ + jr] == 0) v = -INFINITY;   // pad mask on key dim j
        smem[wv][h * 17 + mr] = v;             // stride 17: conflict-free writes
    }
    __syncthreads();

    // coalesced stores: lane pair covers 16 contiguous j per head
    const int h2    = lane >> 1;
    const int half2 = lane & 1;
    float v0[4], v1[4];
    #pragma unroll
    for (int t = 0; t < 4; ++t) v0[t] = smem[wv][h2 * 17 + half2 * 8 + t];
    #pragma unroll
    for (int t = 0; t < 4; ++t) v1[t] = smem[wv][h2 * 17 + half2 * 8 + 4 + t];

    float* outp = out_ap + ((((size_t)b * NHEAD + h2) * LSEQ + i) * LSEQ + j0 + half2 * 8);
    *reinterpret_cast<float4*>(outp)     = make_float4(v0[0], v0[1], v0[2], v0[3]);
    *reinterpret_cast<float4*>(outp + 4) = make_float4(v1[0], v1[1], v1[2], v1[3]);
}

// ---------------------------------------------------------------------------
extern "C" void kernel_launch(void* const* d_in, const int* in_sizes, int n_in,
                              void* d_out, int out_size, void* d_ws, size_t ws_size,
                              hipStream_t stream) {
    (void)in_sizes; (void)n_in; (void)d_ws; (void)ws_size; (void)out_size;

    const int*   atoms     = (const int*)  d_in[0];
    const int*   chirals   = (const int*)  d_in[1];
    const float* coords    = (const float*)d_in[2];
    const int*   bonds     = (const int*)  d_in[3];
    const float* atype_emb = (const float*)d_in[4];
    const float* chiral_emb= (const float*)d_in[5];
    const float* apair_w   = (const float*)d_in[6];
    const float* apair_b   = (const float*)d_in[7];
    const float* means     = (const float*)d_in[8];
    const float* stds      = (const float*)d_in[9];
    const float* bond_emb  = (const float*)d_in[10];
    const float* lin_w     = (const float*)d_in[11];
    const float* lin_b     = (const float*)d_in[12];

    float* out_emb = (float*)d_out;                                 // [L,B,512]
    float* out_ap  = out_emb + (size_t)LSEQ * BSZ * DMODEL;         // [B,16,L,L]

    atoms_emb_kernel<<<BSZ * LSEQ, DMODEL / 4, 0, stream>>>(
        atoms, chirals, atype_emb, chiral_emb, out_emb);

    const int tiles  = BSZ * LSEQ * JTILES;     // 36864 wave-tiles
    const int blocks = tiles / WPB;             // 4608 blocks of 256 threads
    pair_bias_kernel<<<blocks, WPB * 32, 0, stream>>>(
        atoms, coords, bonds, apair_w, apair_b, means, stds,
        bond_emb, lin_w, lin_b, out_ap);
}